// MAT_60421599920298
// MI455X (gfx1250) — compile-verified
//
#include <hip/hip_runtime.h>
#include <hip/hip_bf16.h>

typedef __attribute__((ext_vector_type(2))) float v2f;
typedef __attribute__((ext_vector_type(8))) float v8f;

#define NPIX   (96 * 96)      // 9216 query positions
#define KDIM   120            // reduction dim of both projections
#define SCALE  0.22360679774997896f   // 20^-0.5

// ---------------------------------------------------------------------------
// GEMM: C[M,N] = A[M,120] @ W[N,120]^T + bias[N]
// One 16x16 output tile per wave via V_WMMA_F32_16X16X4_F32 (fp32 in/out).
// Block = 256 threads = 8 waves; each wave takes one M-tile, all waves share
// one N-tile whose 120x16 weight panel is staged in LDS.
// Grid: x = M/128 blocks, y = ceil(N/16) tiles.
// ---------------------------------------------------------------------------
__global__ __launch_bounds__(256)
void gemm_wmma_kernel(const float* __restrict__ A,
                      const float* __restrict__ W,
                      const float* __restrict__ bias,
                      float* __restrict__ C,
                      int N)
{
    __shared__ float Bs[KDIM * 16];   // Bs[k*16 + col] = W[(n0+col)*120 + k]

    const int tid  = threadIdx.x;
    const int wave = tid >> 5;
    const int lane = tid & 31;
    const int half = lane >> 4;       // 0: lanes 0-15, 1: lanes 16-31
    const int l16  = lane & 15;
    const int kp   = half << 1;       // K-pair offset within a 4-step

    const int n0 = blockIdx.y * 16;
    const int m0 = (blockIdx.x * 8 + wave) * 16;

    // Stage the 120x16 weight panel (transposed) into LDS, zero-fill OOB cols.
    for (int idx = tid; idx < KDIM * 16; idx += 256) {
        const int kk  = idx >> 4;
        const int col = idx & 15;
        const int n   = n0 + col;
        Bs[idx] = (n < N) ? W[n * KDIM + kk] : 0.0f;
    }
    __syncthreads();

    // A fragment row for this lane (ISA 32-bit A 16x4 layout).
    const float* __restrict__ ap = A + (m0 + l16) * KDIM;

    v8f acc = {};
    #pragma unroll 6
    for (int k = 0; k < KDIM; k += 4) {
        v2f a, b;
        a.x = ap[k + kp];
        a.y = ap[k + kp + 1];
        b.x = Bs[(k + kp)     * 16 + l16];
        b.y = Bs[(k + kp + 1) * 16 + l16];
        // D = A(16x4) * B(4x16) + C   (fp32 WMMA, 8-arg form)
        acc = __builtin_amdgcn_wmma_f32_16x16x4_f32(
            /*neg_a=*/false, a, /*neg_b=*/false, b,
            /*c_mod=*/(short)0, acc, /*reuse_a=*/false, /*reuse_b=*/false);
    }

    // Store (C/D layout: VGPR v -> row m0 + v + 8*half, col = lane&15).
    const int n = n0 + l16;
    if (n < N) {
        const float bv = bias[n];
        #pragma unroll
        for (int v = 0; v < 8; ++v) {
            const int r = m0 + v + 8 * half;
            C[r * N + n] = acc[v] + bv;
        }
    }
}

// ---------------------------------------------------------------------------
// Neighborhood attention, dilation 1, one thread per (pixel, global head).
// Online (flash) softmax over the KxK window; q and o live in registers.
// qkv layout per pixel (360 ch): group g in [0,3): [q h0|q h1|k h0|k h1|v h0|v h1]*20
// y layout per pixel (120 ch): g*40 + h*20 + d
// ---------------------------------------------------------------------------
__global__ __launch_bounds__(256)
void na2d_kernel(const float* __restrict__ qkv,
                 const float* __restrict__ rpb0,
                 const float* __restrict__ rpb1,
                 const float* __restrict__ rpb2,
                 float* __restrict__ y)
{
    const int gid = blockIdx.x * blockDim.x + threadIdx.x;
    if (gid >= NPIX * 6) return;

    const int pix = gid % NPIX;
    const int hg  = gid / NPIX;     // 0..5 global head
    const int g   = hg >> 1;        // kernel-size group
    const int h   = hg & 1;         // head within group
    const int i   = pix / 96;
    const int j   = pix % 96;

    const int Kw = (g == 0) ? 7 : ((g == 1) ? 9 : 11);
    const int NS = Kw >> 1;
    const int RP = 2 * Kw - 1;
    const float* __restrict__ rpb = (g == 0) ? rpb0 : ((g == 1) ? rpb1 : rpb2);
    const float* __restrict__ rpbh = rpb + h * RP * RP;

    // Clamped window start (d==1 path of _win_start).
    int rs = i - NS; if (rs < 0) rs = 0; if (rs > 96 - Kw) rs = 96 - Kw;
    int cs = j - NS; if (cs < 0) cs = 0; if (cs > 96 - Kw) cs = 96 - Kw;
    // Bias window start (d==1 path of _pb_start).
    const int pr = (i < NS) ? (Kw - 1 - i) : ((i + NS >= 96) ? (96 - 1 - i) : NS);
    const int pc = (j < NS) ? (Kw - 1 - j) : ((j + NS >= 96) ? (96 - 1 - j) : NS);

    // Pre-scaled query in registers.
    const float* __restrict__ qp = qkv + pix * 360 + g * 120 + h * 20;
    float q[20];
    #pragma unroll
    for (int d = 0; d < 20; ++d) q[d] = qp[d] * SCALE;

    float m = -1.0e30f, s = 0.0f;
    float o[20];
    #pragma unroll
    for (int d = 0; d < 20; ++d) o[d] = 0.0f;

    for (int kr = 0; kr < Kw; ++kr) {
        const int nrow = rs + kr;
        const float* __restrict__ rb = rpbh + (pr + kr) * RP + pc;
        for (int kc = 0; kc < Kw; ++kc) {
            const int npix = nrow * 96 + (cs + kc);
            const float* __restrict__ kp2 = qkv + npix * 360 + g * 120 + 40 + h * 20;
            const float* __restrict__ vp  = kp2 + 40;

            float logit = rb[kc];
            #pragma unroll
            for (int d = 0; d < 20; ++d) logit = fmaf(q[d], kp2[d], logit);

            const float mn = fmaxf(m, logit);
            const float c  = __expf(m - mn);      // rescale of old state
            const float p  = __expf(logit - mn);  // weight of this neighbor
            s = s * c + p;
            #pragma unroll
            for (int d = 0; d < 20; ++d) o[d] = fmaf(o[d], c, p * vp[d]);
            m = mn;
        }
    }

    const float inv = 1.0f / s;
    float* __restrict__ yp = y + pix * 120 + g * 40 + h * 20;
    #pragma unroll
    for (int d = 0; d < 20; ++d) yp[d] = o[d] * inv;
}

// ---------------------------------------------------------------------------
extern "C" void kernel_launch(void* const* d_in, const int* in_sizes, int n_in,
                              void* d_out, int out_size, void* d_ws, size_t ws_size,
                              hipStream_t stream)
{
    const float* x      = (const float*)d_in[0];
    const float* qkv_w  = (const float*)d_in[1];
    const float* qkv_b  = (const float*)d_in[2];
    const float* rpb0   = (const float*)d_in[3];
    const float* rpb1   = (const float*)d_in[4];
    const float* rpb2   = (const float*)d_in[5];
    const float* proj_w = (const float*)d_in[6];
    const float* proj_b = (const float*)d_in[7];
    float* out = (float*)d_out;

    float* qkv = (float*)d_ws;             // 9216 * 360 floats (13.3 MB)
    float* y   = qkv + (size_t)NPIX * 360; // 9216 * 120 floats (4.4 MB)

    // 1) qkv = x @ qkv_w^T + qkv_b   (M=9216, N=360, K=120)
    gemm_wmma_kernel<<<dim3(NPIX / 128, 23), 256, 0, stream>>>(x, qkv_w, qkv_b, qkv, 360);

    // 2) neighborhood attention per (pixel, head) -> y [9216,120]
    na2d_kernel<<<dim3((NPIX * 6) / 256), 256, 0, stream>>>(qkv, rpb0, rpb1, rpb2, y);

    // 3) out = y @ proj_w^T + proj_b (M=9216, N=120, K=120)
    gemm_wmma_kernel<<<dim3(NPIX / 128, 8), 256, 0, stream>>>(y, proj_w, proj_b, out, 120);
}